// DomainMix_1992864825358
// MI455X (gfx1250) — compile-verified
//
#include <hip/hip_runtime.h>
#include <math.h>

// Problem constants (from reference)
#define Bc   64
#define Sc   129
#define Fc   768
#define Dc   4
#define SFc  (Sc * Fc)        // 99072
#define BSFc (Bc * SFc)       // 6340608
#define Nr   (3 * Bc)         // 192 rows in xe
#define MOMc 0.9f
#define EPSc 1e-6f

typedef __attribute__((ext_vector_type(2))) float v2f;
typedef __attribute__((ext_vector_type(8))) float v8f;

// ---------------------------------------------------------------------------
// Kernel 1: per-domain segment stats -> new_mean / new_var  (D x F)
// grid = 3 x 256 threads; thread owns one feature column f.
// ---------------------------------------------------------------------------
__global__ void stats_kernel(const float* __restrict__ x,
                             const int* __restrict__ domain,
                             const float* __restrict__ mean_buf,
                             const float* __restrict__ var_buf,
                             float* __restrict__ out_mean,
                             float* __restrict__ out_var) {
  __shared__ int sdom[Bc];
  const int tid = threadIdx.x;
  if (tid < Bc) sdom[tid] = domain[tid];
  __syncthreads();

  const int f = blockIdx.x * blockDim.x + tid;   // 0..767

  int cb[Dc] = {0, 0, 0, 0};
  for (int b = 0; b < Bc; ++b) cb[sdom[b]]++;

  float s1[Dc] = {0.f, 0.f, 0.f, 0.f};
  float s2[Dc] = {0.f, 0.f, 0.f, 0.f};
  for (int b = 0; b < Bc; ++b) {
    const int d = sdom[b];
    const float* row = x + (size_t)b * SFc + f;
    float a1 = 0.f, a2 = 0.f;
    for (int s = 0; s < Sc; ++s) {
      float v = row[(size_t)s * Fc];
      a1 += v;
      a2 += v * v;
    }
    s1[d] += a1;
    s2[d] += a2;
  }

  #pragma unroll
  for (int d = 0; d < Dc; ++d) {
    const float n   = (float)cb[d] * (float)Sc;
    const float mu  = s1[d] / fmaxf(n, 1.f);
    const float var = (s2[d] - n * mu * mu) / fmaxf(n - 1.f, 1.f);
    const int   idx = d * Fc + f;
    const float mb  = mean_buf[idx];
    const float vb  = var_buf[idx];
    const bool  present = (cb[d] > 0);
    out_mean[idx] = present ? (MOMc * mb + (1.f - MOMc) * mu)  : mb;
    out_var[idx]  = present ? (MOMc * vb + (1.f - MOMc) * var) : vb;
  }
}

// ---------------------------------------------------------------------------
// Kernel 2: instance norm + domain mix + hallucinated samples + row |.|^2
// grid = 64 blocks (one per b) x 256 threads.
// ---------------------------------------------------------------------------
__global__ void mix_kernel(const float* __restrict__ x,
                           const float* __restrict__ lmda,
                           const float* __restrict__ hg_noise,
                           const int* __restrict__ domain,
                           const int* __restrict__ d_rand,
                           const float* __restrict__ nmean,
                           const float* __restrict__ nvar,
                           float* __restrict__ xmix,
                           float* __restrict__ hg,
                           float* __restrict__ sq) {
  __shared__ float smu[Fc], sistd[Fc];
  __shared__ float ssc_ds[Fc], smn_ds[Fc], ssc_do[Fc], smn_do[Fc];
  __shared__ float r0[256], r1[256], r2[256];

  const int b   = blockIdx.x;
  const int tid = threadIdx.x;
  const int dd  = domain[b];
  const int ds  = (dd + d_rand[b]) % Dc;

  // Phase A: per-(b,f) stats over S, plus broadcast scale/shift vectors
  for (int f = tid; f < Fc; f += 256) {
    const float* col = x + (size_t)b * SFc + f;
    float a1 = 0.f, a2 = 0.f;
    for (int s = 0; s < Sc; ++s) {
      float v = col[(size_t)s * Fc];
      a1 += v;
      a2 += v * v;
    }
    const float mu  = a1 / (float)Sc;
    const float var = (a2 - (float)Sc * mu * mu) / (float)(Sc - 1);
    smu[f]    = mu;
    sistd[f]  = rsqrtf(var + EPSc);
    ssc_ds[f] = sqrtf(nvar[ds * Fc + f] + EPSc);
    smn_ds[f] = nmean[ds * Fc + f];
    ssc_do[f] = sqrtf(nvar[dd * Fc + f] + EPSc);
    smn_do[f] = nmean[dd * Fc + f];
  }
  __syncthreads();

  // Phase B: produce x_mix and hg, accumulate row sums of squares
  const float lm = lmda[b];
  float sqx = 0.f, sqm = 0.f, sqh = 0.f;
  const size_t base = (size_t)b * SFc;
  for (int idx = tid; idx < SFc; idx += 256) {
    const int f = idx % Fc;
    const float v  = x[base + idx];
    const float xn = (v - smu[f]) * sistd[f];
    const float xw = xn * ssc_ds[f] + smn_ds[f];
    const float xm = lm * v + (1.f - lm) * xw;
    const float h  = smn_do[f] + ssc_do[f] * hg_noise[base + idx];
    xmix[base + idx] = xm;
    hg[base + idx]   = h;
    sqx += v * v;
    sqm += xm * xm;
    sqh += h * h;
  }
  r0[tid] = sqx; r1[tid] = sqm; r2[tid] = sqh;
  __syncthreads();
  for (int off = 128; off > 0; off >>= 1) {
    if (tid < off) {
      r0[tid] += r0[tid + off];
      r1[tid] += r1[tid + off];
      r2[tid] += r2[tid + off];
    }
    __syncthreads();
  }
  if (tid == 0) {
    sq[b]          = r0[0];
    sq[Bc + b]     = r1[0];
    sq[2 * Bc + b] = r2[0];
  }
}

// ---------------------------------------------------------------------------
// Kernel 3: Gram matrix G = Xe @ Xe^T via V_WMMA_F32_16X16X4_F32.
// grid = 12 x 12 tiles, 256 threads (8 waves); K=99072 split 8 ways.
// Per lane, the 16x4 fp32 A fragment is a contiguous float2 at
// k + (lane<16 ? 0 : 2) of row I*16+(lane&15); B fragment identical with
// rows J*16+(lane&15). Inner loop: unroll x4, two accumulator chains, stepped
// pointers with immediate offsets -> 8 b64 loads issued ahead of partial
// loadcnt waits.
// ---------------------------------------------------------------------------
#define GRAM_WAVES 8

__global__ __launch_bounds__(32 * GRAM_WAVES)
void gram_kernel(const float* __restrict__ X,
                 const float* __restrict__ M,
                 const float* __restrict__ H,
                 float* __restrict__ G) {
  const int lane = threadIdx.x & 31;
  const int wave = threadIdx.x >> 5;
  const int I = blockIdx.y, J = blockIdx.x;
  const int m = lane & 15;

  const int rA = I * 16 + m;
  const int rB = J * 16 + m;
  const float* Ap = (rA < Bc) ? (X + (size_t)rA * SFc)
                  : (rA < 2 * Bc) ? (M + (size_t)(rA - Bc) * SFc)
                                  : (H + (size_t)(rA - 2 * Bc) * SFc);
  const float* Bp = (rB < Bc) ? (X + (size_t)rB * SFc)
                  : (rB < 2 * Bc) ? (M + (size_t)(rB - Bc) * SFc)
                                  : (H + (size_t)(rB - 2 * Bc) * SFc);
  const int koff = (lane < 16) ? 0 : 2;

  // wave w owns k in [w*16, w*16+16) then strides by GRAM_WAVES*16 = 128
  const float* ap = Ap + wave * 16 + koff;
  const float* bp = Bp + wave * 16 + koff;

  v8f acc0 = {0.f, 0.f, 0.f, 0.f, 0.f, 0.f, 0.f, 0.f};
  v8f acc1 = {0.f, 0.f, 0.f, 0.f, 0.f, 0.f, 0.f, 0.f};

  const int iters = SFc / (GRAM_WAVES * 16);   // 99072 / 128 = 774
  for (int it = 0; it < iters; ++it) {
    v2f a0 = *(const v2f*)(ap + 0);
    v2f b0 = *(const v2f*)(bp + 0);
    v2f a1 = *(const v2f*)(ap + 4);
    v2f b1 = *(const v2f*)(bp + 4);
    v2f a2 = *(const v2f*)(ap + 8);
    v2f b2 = *(const v2f*)(bp + 8);
    v2f a3 = *(const v2f*)(ap + 12);
    v2f b3 = *(const v2f*)(bp + 12);
    acc0 = __builtin_amdgcn_wmma_f32_16x16x4_f32(false, a0, false, b0,
                                                 (short)0, acc0, false, false);
    acc1 = __builtin_amdgcn_wmma_f32_16x16x4_f32(false, a1, false, b1,
                                                 (short)0, acc1, false, false);
    acc0 = __builtin_amdgcn_wmma_f32_16x16x4_f32(false, a2, false, b2,
                                                 (short)0, acc0, false, false);
    acc1 = __builtin_amdgcn_wmma_f32_16x16x4_f32(false, a3, false, b3,
                                                 (short)0, acc1, false, false);
    ap += GRAM_WAVES * 16;
    bp += GRAM_WAVES * 16;
  }

  __shared__ float red[GRAM_WAVES][8][32];
  #pragma unroll
  for (int i = 0; i < 8; ++i) red[wave][i][lane] = acc0[i] + acc1[i];
  __syncthreads();

  if (wave == 0) {
    #pragma unroll
    for (int i = 0; i < 8; ++i) {
      float v = 0.f;
      #pragma unroll
      for (int w = 0; w < GRAM_WAVES; ++w) v += red[w][i][lane];
      const int Mr = (lane < 16) ? i : i + 8;       // C/D VGPR layout
      G[(size_t)(I * 16 + Mr) * Nr + (J * 16 + m)] = v;
    }
  }
}

// ---------------------------------------------------------------------------
// Kernel 4: hardest-positive / hardest-negative triplet loss. 1 block.
// ---------------------------------------------------------------------------
__global__ void loss_kernel(const float* __restrict__ G,
                            const float* __restrict__ sq,
                            const int* __restrict__ labels,
                            float* __restrict__ loss_out) {
  __shared__ float ssq[Nr];
  __shared__ int   sl[Nr];
  __shared__ float red[256];
  const int tid = threadIdx.x;

  if (tid < Nr) {
    ssq[tid] = sq[tid];
    sl[tid]  = (tid < 2 * Bc) ? labels[tid % Bc] : -1;
  }
  __syncthreads();

  float contrib = 0.f;
  if (tid < Nr) {
    const int   li  = sl[tid];
    const float sqi = ssq[tid];
    float ap = -1e30f, an = 1e30f;
    for (int j = 0; j < Nr; ++j) {
      const float d2 = sqi + ssq[j] - 2.f * G[(size_t)tid * Nr + j];
      const float dist = sqrtf(fmaxf(d2, 1e-12f));
      if (sl[j] == li) ap = fmaxf(ap, dist);
      else             an = fminf(an, dist);
    }
    const float t = ap - an;
    contrib = (t > 20.f) ? t : log1pf(expf(t));   // stable softplus
  }
  red[tid] = contrib;
  __syncthreads();
  for (int off = 128; off > 0; off >>= 1) {
    if (tid < off) red[tid] += red[tid + off];
    __syncthreads();
  }
  if (tid == 0) loss_out[0] = red[0] / (float)Nr;
}

// ---------------------------------------------------------------------------
extern "C" void kernel_launch(void* const* d_in, const int* in_sizes, int n_in,
                              void* d_out, int out_size, void* d_ws, size_t ws_size,
                              hipStream_t stream) {
  const float* x        = (const float*)d_in[0];
  const float* lmda     = (const float*)d_in[1];
  const float* mean_buf = (const float*)d_in[2];
  const float* var_buf  = (const float*)d_in[3];
  const float* hg_noise = (const float*)d_in[4];
  const int*   labels   = (const int*)d_in[5];
  const int*   domain   = (const int*)d_in[6];
  const int*   d_rand   = (const int*)d_in[7];

  float* out = (float*)d_out;
  // d_out layout: x_mix [0, BSF) | loss [BSF] | new_mean (D*F) | new_var (D*F)
  float* out_xmix = out;
  float* out_loss = out + BSFc;
  float* out_mean = out + BSFc + 1;
  float* out_var  = out + BSFc + 1 + Dc * Fc;

  // ws layout: hg (B*S*F) | sq (192) | G (192*192)
  float* ws  = (float*)d_ws;
  float* hg  = ws;
  float* sq  = ws + BSFc;
  float* G   = ws + BSFc + Nr;

  stats_kernel<<<3, 256, 0, stream>>>(x, domain, mean_buf, var_buf,
                                      out_mean, out_var);
  mix_kernel<<<Bc, 256, 0, stream>>>(x, lmda, hg_noise, domain, d_rand,
                                     out_mean, out_var, out_xmix, hg, sq);
  dim3 gg(Nr / 16, Nr / 16);
  gram_kernel<<<gg, 32 * GRAM_WAVES, 0, stream>>>(x, out_xmix, hg, G);
  loss_kernel<<<1, 256, 0, stream>>>(G, sq, labels, out_loss);
}